// XAIDiffNet_8942121911085
// MI455X (gfx1250) — compile-verified
//
#include <hip/hip_runtime.h>
#include <hip/hip_bf16.h>
#include <math.h>

// ---------------------------------------------------------------------------
// XAIDiffNet on MI455X (gfx1250), fp32 via V_WMMA_F32_16X16X4_F32 with all
// GEMM tile staging done by the Tensor Data Mover (TENSOR_LOAD_TO_LDS),
// double-buffered in LDS so DMA overlaps the WMMA stream.
//
// Sizes: B=16, T=12, P=12, N=512, U=64, L=2, K=2, IN_DIM=1.
// Internal layout: node-major [N][B][d] so that
//   - graph propagation is   Pcat[2N,N] @ X[N, B*128]      (one GEMM)
//   - weight projection is   X[N*B,128] @ W[128,dout]      (same GEMM kernel)
//
// Input flattening assumption (top-level dict in insertion order, params
// pytree in jax order = dict keys sorted):
//   [0]=x [1]=adj [2]=mask1 [3]=mask2 [4]=tmask
//   [5..28]  dec: cell{0,1} x {c1,c2,g1,g2} x {Wb,Wf,b}
//   [29..52] enc: same
//   [53]=in_fc.W [54]=in_fc.b [55]=out_fc.W [56]=out_fc.b
//
// Workspace requirement: ~44 MB (see offsets below).
// ---------------------------------------------------------------------------

#define NN 512
#define BB 16
#define UU 64
#define CD 128            // combined dim = 2U
#define BD (BB*CD)        // 2048 columns in propagation GEMM
#define MROWS (NN*BB)     // 8192 rows in weight GEMMs

typedef float v2f __attribute__((ext_vector_type(2)));
typedef float v8f __attribute__((ext_vector_type(8)));
typedef unsigned int u32x4 __attribute__((ext_vector_type(4)));
typedef int i32x4 __attribute__((ext_vector_type(4)));
typedef int i32x8 __attribute__((ext_vector_type(8)));

// ------------------------- TDM 2D tile load --------------------------------
// Issue one Tensor-DMA: load a (tile_h x tile_w) fp32 tile (row stride
// `row_stride` elements) from `gaddr` into LDS at byte offset `ldsOff`.
// `clip_w` = valid elements from tile start along X: X >= clip_w reads zero
// (hardware OOB fill). If `pad`, insert 1 dword of LDS padding after every
// 16 dwords stored (ISA: pad_interval=3 -> 16 dwords, pad_amount=0 -> 1).
// D# packing per CDNA5 ISA 8.3/8.4 (group0: count/lds/global/type,
// group1: data_size, dims, tile dims, 48-bit stride). Groups 2/3 unused (2D);
// this toolchain's 6-arg builtin takes an extra (unused here) i32x8 group.
__device__ __attribute__((always_inline)) void tdm_load_2d(
    unsigned ldsOff, const float* gaddr,
    unsigned tile_w, unsigned tile_h,
    unsigned clip_w, unsigned long long row_stride, bool pad)
{
    unsigned long long ga = (unsigned long long)gaddr;
    u32x4 g0;
    g0[0] = 1u;                                  // count=1, user mode, no gather
    g0[1] = ldsOff;                              // lds_addr (bytes)
    g0[2] = (unsigned)ga;                        // global_addr[31:0]
    g0[3] = (unsigned)(ga >> 32) | (2u << 30);   // global_addr[56:32] | type=2

    unsigned d0 = (2u << 16);                    // data_size = 4 bytes
    if (pad) d0 |= (1u << 20) | (3u << 22);      // pad_enable, every 16 dw, +1 dw

    unsigned long long td0 = clip_w;             // tensor_dim0 (OOB clip along X)
    unsigned td1 = 0x40000000u;                  // tensor_dim1: effectively inf
    unsigned long long s0 = row_stride;          // tensor_dim0_stride (elements)

    i32x8 g1;
    g1[0] = (int)d0;                                            // mask|size|pad
    g1[1] = (int)((unsigned)(td0 & 0xffffu) << 16);             // dim0 lo16 @ [63:48]
    g1[2] = (int)(((td0 >> 16) & 0xffffu) | ((td1 & 0xffffu) << 16)); // dim0 hi | dim1 lo
    g1[3] = (int)(((td1 >> 16) & 0xffffu) | (tile_w << 16));    // dim1 hi | tile_dim0
    g1[4] = (int)(tile_h & 0xffffu);                            // tile_dim1 | tile_dim2=0
    g1[5] = (int)(unsigned)(s0 & 0xffffffffu);                  // stride0[31:0]
    g1[6] = (int)(unsigned)((s0 >> 32) & 0xffffu);              // stride0[47:32]
    g1[7] = 0;                                                  // stride1 (unused, 2D)

    i32x4 gz = {0, 0, 0, 0};
    i32x8 gz8 = {0, 0, 0, 0, 0, 0, 0, 0};
    __builtin_amdgcn_tensor_load_to_lds(g0, g1, gz, gz, gz8, 0);
}

// ------------------------- WMMA GEMM ---------------------------------------
// C[M,Nc] (optionally +=) A[M,K] @ B[K,Nc]  (+ bias[col]) (+ activation)
// flags: bit0 = accumulate into existing C; bits[2:1]: 0 none, 1 sigmoid, 2 tanh
#define TBM 64
#define TBN 128
#define TBK 16
#define AROW 17                 // A row stride in LDS (16 + 1 TDM pad dword)
#define ASZ  (TBM*AROW + 4)     // slack for trailing pad dword

__global__ __launch_bounds__(256) void k_gemm(
    const float* __restrict__ A, const float* __restrict__ B,
    float* __restrict__ C, const float* __restrict__ bias,
    int M, int Nc, int Kd, int flags)
{
    // Double-buffered tiles, filled exclusively by the Tensor Data Mover.
    __shared__ float As[2][ASZ];        // A tile, row-major 64x16, stride 17
    __shared__ float Bs[2][TBK * TBN];  // B tile, row-major 16x128, packed

    const int tid  = threadIdx.x;
    const int row0 = blockIdx.y * TBM;
    const int col0 = blockIdx.x * TBN;
    const int wave = tid >> 5;
    const int lane = tid & 31;
    const int half = lane >> 4;          // 0: lanes 0-15, 1: lanes 16-31
    const int mr   = lane & 15;
    const int wr   = (wave >> 2) * 32;   // wave row offset (0/32)
    const int wc   = (wave & 3) * 32;    // wave col offset (0..96)

    // LDS byte offsets (low 32 bits of the flat LDS address = in-group offset)
    const unsigned baseA = (unsigned)(unsigned long long)(void*)&As[0][0];
    const unsigned baseB = (unsigned)(unsigned long long)(void*)&Bs[0][0];
    const unsigned strA  = ASZ * 4;
    const unsigned strB  = TBK * TBN * 4;

    v8f acc[2][2] = {};

    // Prologue: wave 0 DMAs the first A/B tile pair.
    if (wave == 0) {
        tdm_load_2d(baseA, A + (size_t)row0 * Kd, TBK, TBM,
                    (unsigned)Kd, (unsigned long long)Kd, true);
        tdm_load_2d(baseB, B + (size_t)col0, TBN, TBK,
                    (unsigned)(Nc - col0), (unsigned long long)Nc, false);
    }

    int buf = 0;
    for (int k0 = 0; k0 < Kd; k0 += TBK) {
        // Prefetch next tile pair into the other buffer, then wait for the
        // current pair (TENSORcnt is in-order per wave: <=2 outstanding means
        // the older pair has landed in LDS).
        if (wave == 0) {
            if (k0 + TBK < Kd) {
                int nb = buf ^ 1;
                tdm_load_2d(baseA + nb * strA,
                            A + (size_t)row0 * Kd + (k0 + TBK), TBK, TBM,
                            (unsigned)(Kd - (k0 + TBK)), (unsigned long long)Kd,
                            true);
                tdm_load_2d(baseB + nb * strB,
                            B + (size_t)(k0 + TBK) * Nc + col0, TBN, TBK,
                            (unsigned)(Nc - col0), (unsigned long long)Nc, false);
                __builtin_amdgcn_s_wait_tensorcnt(2);
            } else {
                __builtin_amdgcn_s_wait_tensorcnt(0);
            }
        }
        __syncthreads();   // tiles in `buf` are visible to all waves

        const float* Af = &As[buf][0];
        const float* Bf = &Bs[buf][0];
        #pragma unroll
        for (int kk = 0; kk < TBK; kk += 4) {
            // A frag 16x4: lane(half,mr) holds A[m][kk+2*half+{0,1}]
            // (stride-17 rows -> conflict-free across the 16 lanes)
            v2f af[2]; v2f bf[2];
            #pragma unroll
            for (int i = 0; i < 2; ++i) {
                int m = wr + i * 16 + mr;
                af[i].x = Af[m * AROW + kk + 2 * half + 0];
                af[i].y = Af[m * AROW + kk + 2 * half + 1];
            }
            // B frag 4x16: lane(half,mr) holds B[kk+2*half+{0,1}][n]
            #pragma unroll
            for (int j = 0; j < 2; ++j) {
                int n = wc + j * 16 + mr;
                bf[j].x = Bf[(kk + 2 * half + 0) * TBN + n];
                bf[j].y = Bf[(kk + 2 * half + 1) * TBN + n];
            }
            #pragma unroll
            for (int i = 0; i < 2; ++i)
                #pragma unroll
                for (int j = 0; j < 2; ++j)
                    acc[i][j] = __builtin_amdgcn_wmma_f32_16x16x4_f32(
                        false, af[i], false, bf[j], (short)0, acc[i][j],
                        false, false);
        }
        __syncthreads();   // everyone done reading `buf` before TDM reuses it
        buf ^= 1;
    }

    const int  act   = (flags >> 1) & 3;
    const bool accum = flags & 1;
    // C/D layout: VGPR v of lane(half,mr) = C[v + 8*half][mr] within 16x16 tile
    #pragma unroll
    for (int i = 0; i < 2; ++i) {
        #pragma unroll
        for (int j = 0; j < 2; ++j) {
            #pragma unroll
            for (int v = 0; v < 8; ++v) {
                int r = row0 + wr + i * 16 + v + 8 * half;
                int c = col0 + wc + j * 16 + mr;
                if (c < Nc) {
                    size_t off = (size_t)r * Nc + c;
                    float val = acc[i][j][v];
                    if (accum) val += C[off];
                    if (bias)  val += bias[c];
                    if (act == 1)      val = 1.0f / (1.0f + expf(-val));
                    else if (act == 2) val = tanhf(val);
                    C[off] = val;
                }
            }
        }
    }
}

// ------------------------- small elementwise kernels -----------------------

__global__ void k_prep_adj(const float* __restrict__ adj,
                           const float* __restrict__ m1,
                           const float* __restrict__ m2,
                           float* __restrict__ A1, float* __restrict__ A2) {
    int i = blockIdx.x * blockDim.x + threadIdx.x;
    if (i >= NN * NN) return;
    int m = i / NN, n = i % NN;
    float e = (m == n) ? 1.0f : 0.0f;
    float a = adj[i];
    A1[i] = a * (tanhf((m1[m * NN + n] + m1[n * NN + m]) * 0.5f) + 1.0f) + e;
    A2[i] = a * (tanhf((m2[m * NN + n] + m2[n * NN + m]) * 0.5f) + 1.0f) + e;
}

__global__ void k_sums(const float* __restrict__ A,
                       float* __restrict__ rs, float* __restrict__ cs) {
    __shared__ float sr[256], sc[256];
    int m = blockIdx.x, t = threadIdx.x;
    float r = 0.f, c = 0.f;
    for (int n = t; n < NN; n += 256) { r += A[m * NN + n]; c += A[n * NN + m]; }
    sr[t] = r; sc[t] = c; __syncthreads();
    for (int s = 128; s > 0; s >>= 1) {
        if (t < s) { sr[t] += sr[t + s]; sc[t] += sc[t + s]; }
        __syncthreads();
    }
    if (t == 0) { rs[m] = sr[0]; cs[m] = sc[0]; }
}

// P[0:N]   = row-normalized A      (P_fwd)
// P[N:2N]  = row-normalized A^T    (P_bwd)
__global__ void k_make_P(const float* __restrict__ A,
                         const float* __restrict__ rs,
                         const float* __restrict__ cs,
                         float* __restrict__ P) {
    int i = blockIdx.x * blockDim.x + threadIdx.x;
    if (i >= NN * NN) return;
    int m = i / NN, n = i % NN;
    float r = rs[m], c = cs[m];
    P[i]           = (r > 0.f) ? A[i] / r : 0.f;
    P[NN * NN + i] = (c > 0.f) ? A[n * NN + m] / c : 0.f;
}

__global__ void k_wsum(const float* __restrict__ Wf, const float* __restrict__ Wb,
                       float* __restrict__ Ws, int n) {
    int i = blockIdx.x * blockDim.x + threadIdx.x;
    if (i < n) Ws[i] = Wf[i] + Wb[i];
}

__global__ void k_zero(float* __restrict__ p, int n) {
    int i = blockIdx.x * blockDim.x + threadIdx.x;
    if (i < n) p[i] = 0.f;
}

// xe[n][b][u] = x[b][t][n][0] * (tanh(tm[t][n])+1) * Win[u] + bin[u]
__global__ void k_embed_enc(const float* __restrict__ x, const float* __restrict__ tm,
                            const float* __restrict__ Win, const float* __restrict__ bin,
                            float* __restrict__ xe, int t) {
    int i = blockIdx.x * blockDim.x + threadIdx.x;
    if (i >= NN * BB * UU) return;
    int u = i & 63, b = (i >> 6) & 15, n = i >> 10;
    float xv = x[(b * 12 + t) * NN + n];
    float sc = tanhf(tm[t * NN + n]) + 1.0f;
    xe[i] = xv * sc * Win[u] + bin[u];
}

// xe[n][b][u] = go[b][n] * Win[u] + bin[u]
__global__ void k_embed_dec(const float* __restrict__ go,
                            const float* __restrict__ Win, const float* __restrict__ bin,
                            float* __restrict__ xe) {
    int i = blockIdx.x * blockDim.x + threadIdx.x;
    if (i >= NN * BB * UU) return;
    int u = i & 63, b = (i >> 6) & 15, n = i >> 10;
    xe[i] = go[b * NN + n] * Win[u] + bin[u];
}

// xh[nb][0:64]=xin, xh[nb][64:128]=h
__global__ void k_concat(const float* __restrict__ xin, const float* __restrict__ h,
                         float* __restrict__ xh) {
    int i = blockIdx.x * blockDim.x + threadIdx.x;
    if (i >= NN * BB * CD) return;
    int d = i & 127, nb = i >> 7;
    xh[i] = (d < UU) ? xin[nb * UU + d] : h[nb * UU + (d - UU)];
}

// xh[nb][0:64]=xin, xh[nb][64:128]= r * h,  r = g[nb][0:64] (sigmoid applied)
__global__ void k_concat_rh(const float* __restrict__ xin, const float* __restrict__ g,
                            const float* __restrict__ h, float* __restrict__ xh) {
    int i = blockIdx.x * blockDim.x + threadIdx.x;
    if (i >= NN * BB * CD) return;
    int d = i & 127, nb = i >> 7;
    xh[i] = (d < UU) ? xin[nb * UU + d]
                     : g[nb * CD + (d - UU)] * h[nb * UU + (d - UU)];
}

// h = (1-u)*h + u*c, u = (g1[.,64+u]+g2[.,64+u])/2, c = (c1+c2)/2
__global__ void k_update(const float* __restrict__ g1, const float* __restrict__ g2,
                         const float* __restrict__ c1, const float* __restrict__ c2,
                         float* __restrict__ h) {
    int i = blockIdx.x * blockDim.x + threadIdx.x;
    if (i >= NN * BB * UU) return;
    int u = i & 63, nb = i >> 6;
    float uu = 0.5f * (g1[nb * CD + UU + u] + g2[nb * CD + UU + u]);
    float cc = 0.5f * (c1[i] + c2[i]);
    h[i] = (1.0f - uu) * h[i] + uu * cc;
}

// out[b][p][n] = dot(h[n][b][:], Wout) + bout ; also go[b][n] = same
__global__ void k_out(const float* __restrict__ h, const float* __restrict__ Wout,
                      const float* __restrict__ bout, float* __restrict__ out,
                      float* __restrict__ go, int p) {
    int i = blockIdx.x * blockDim.x + threadIdx.x;
    if (i >= BB * NN) return;
    int b = i >> 9, n = i & 511;
    const float* hp = &h[(n * BB + b) * UU];
    float s = bout[0];
    #pragma unroll
    for (int u = 0; u < UU; ++u) s += hp[u] * Wout[u];
    out[(b * 12 + p) * NN + n] = s;
    go[i] = s;
}

// ------------------------- host-side orchestration -------------------------

struct Dgc { const float* Wf1; const float* Wb1; const float* b; const float* Wsum; };

static inline void gemm(const float* A, const float* B, float* C, const float* bias,
                        int M, int Nc, int Kd, int flags, hipStream_t s) {
    dim3 g((Nc + TBN - 1) / TBN, M / TBM);
    k_gemm<<<g, 256, 0, s>>>(A, B, C, bias, M, Nc, Kd, flags);
}

// out[N*B, dout] = act( xin@Wsum + (Pf@xin)@Wf1 + (Pb@xin)@Wb1 + b )
static void run_dgc(const float* xin, const float* P, const Dgc& p,
                    float* t, float* out, int dout, int act, hipStream_t s) {
    gemm(P, xin, t, nullptr, 2 * NN, BD, NN, 0, s);          // t = [Pf;Pb] @ X
    gemm(xin, p.Wsum, out, p.b, MROWS, dout, CD, 0, s);      // out = X@Wsum + b
    gemm(t, p.Wf1, out, nullptr, MROWS, dout, CD, 1, s);     // out += tf@Wf1
    gemm(t + (size_t)NN * BD, p.Wb1, out, nullptr, MROWS, dout, CD,
         1 | (act << 1), s);                                  // out += tb@Wb1, act
}

// workspace offsets (floats)
static const size_t O_A1 = 0;
static const size_t O_A2 = O_A1 + (size_t)NN * NN;
static const size_t O_RS = O_A2 + (size_t)NN * NN;          // rs1,cs1,rs2,cs2
static const size_t O_P1 = O_RS + 4 * NN;
static const size_t O_P2 = O_P1 + 2 * (size_t)NN * NN;
static const size_t O_WS = O_P2 + 2 * (size_t)NN * NN;      // 16 x (128*128)
static const size_t O_H  = O_WS + 16 * 16384;
static const size_t O_XE = O_H  + 2 * (size_t)NN * BB * UU;
static const size_t O_XH = O_XE + (size_t)NN * BB * UU;
static const size_t O_T  = O_XH + (size_t)NN * BB * CD;
static const size_t O_G1 = O_T  + 2 * (size_t)NN * BD;
static const size_t O_G2 = O_G1 + (size_t)NN * BB * CD;
static const size_t O_XR = O_G2 + (size_t)NN * BB * CD;
static const size_t O_C1 = O_XR + (size_t)NN * BB * CD;
static const size_t O_C2 = O_C1 + (size_t)NN * BB * UU;
static const size_t O_GO = O_C2 + (size_t)NN * BB * UU;     // total ~10.76M floats

extern "C" void kernel_launch(void* const* d_in, const int* in_sizes, int n_in,
                              void* d_out, int out_size, void* d_ws, size_t ws_size,
                              hipStream_t stream) {
    (void)in_sizes; (void)n_in; (void)out_size; (void)ws_size;

    const float* x   = (const float*)d_in[0];
    const float* adj = (const float*)d_in[1];
    const float* m1  = (const float*)d_in[2];
    const float* m2  = (const float*)d_in[3];
    const float* tm  = (const float*)d_in[4];
    const float* Win  = (const float*)d_in[53];
    const float* bin  = (const float*)d_in[54];
    const float* Wout = (const float*)d_in[55];
    const float* bout = (const float*)d_in[56];

    float* ws = (float*)d_ws;
    float* A1 = ws + O_A1;  float* A2 = ws + O_A2;
    float* rs1 = ws + O_RS; float* cs1 = rs1 + NN;
    float* rs2 = cs1 + NN;  float* cs2 = rs2 + NN;
    float* P1 = ws + O_P1;  float* P2 = ws + O_P2;
    float* wsum = ws + O_WS;
    float* H  = ws + O_H;           // [L][N][B][U]
    float* XE = ws + O_XE;
    float* XH = ws + O_XH;
    float* T  = ws + O_T;
    float* G1 = ws + O_G1;
    float* G2 = ws + O_G2;
    float* XR = ws + O_XR;
    float* C1 = ws + O_C1;
    float* C2 = ws + O_C2;
    float* GO = ws + O_GO;
    float* out = (float*)d_out;

    // ---- build dgc param tables: [grp 0=dec,1=enc][cell][which 0=g1,1=c1,2=g2,3=c2]
    Dgc dgc[2][2][4];
    {
        // in-cell (sorted-key) offsets: c1=+0, c2=+3, g1=+6, g2=+9; each {Wb,Wf,b}
        const int off_which[4] = {6, 0, 9, 3};   // g1, c1, g2, c2
        const int dout_which[4] = {2 * UU, UU, 2 * UU, UU};
        int set = 0;
        for (int grp = 0; grp < 2; ++grp) {
            for (int c = 0; c < 2; ++c) {
                int cb = 5 + grp * 24 + c * 12;
                for (int w = 0; w < 4; ++w) {
                    int base = cb + off_which[w];
                    int dout = dout_which[w];
                    const float* Wb = (const float*)d_in[base + 0];
                    const float* Wf = (const float*)d_in[base + 1];
                    const float* b  = (const float*)d_in[base + 2];
                    Dgc d;
                    d.Wf1 = Wf + CD * dout;
                    d.Wb1 = Wb + CD * dout;
                    d.b   = b;
                    d.Wsum = wsum + (size_t)set * 16384;
                    dgc[grp][c][w] = d;
                    int n = CD * dout;
                    k_wsum<<<(n + 255) / 256, 256, 0, stream>>>(Wf, Wb,
                        wsum + (size_t)set * 16384, n);
                    ++set;
                }
            }
        }
    }

    // ---- supports
    k_prep_adj<<<(NN * NN + 255) / 256, 256, 0, stream>>>(adj, m1, m2, A1, A2);
    k_sums<<<NN, 256, 0, stream>>>(A1, rs1, cs1);
    k_sums<<<NN, 256, 0, stream>>>(A2, rs2, cs2);
    k_make_P<<<(NN * NN + 255) / 256, 256, 0, stream>>>(A1, rs1, cs1, P1);
    k_make_P<<<(NN * NN + 255) / 256, 256, 0, stream>>>(A2, rs2, cs2, P2);

    // ---- init state
    k_zero<<<(2 * NN * BB * UU + 255) / 256, 256, 0, stream>>>(H, 2 * NN * BB * UU);
    k_zero<<<(BB * NN + 255) / 256, 256, 0, stream>>>(GO, BB * NN);

    const int nElemCD = NN * BB * CD;   // 1,048,576
    const int nElemU  = NN * BB * UU;   //   524,288

    // ---- 24 recurrent steps
    for (int step = 0; step < 24; ++step) {
        const bool is_enc = step < 12;
        const int grp = is_enc ? 1 : 0;   // d_in layout: dec first, enc second
        if (is_enc) {
            k_embed_enc<<<(nElemU + 255) / 256, 256, 0, stream>>>(x, tm, Win, bin,
                                                                  XE, step);
        } else {
            k_embed_dec<<<(nElemU + 255) / 256, 256, 0, stream>>>(GO, Win, bin, XE);
        }
        for (int layer = 0; layer < 2; ++layer) {
            const float* h_in = (layer == 0) ? XE : H;       // H[0] after its update
            float* h = H + (size_t)layer * nElemU;
            Dgc g1 = dgc[grp][layer][0], c1 = dgc[grp][layer][1];
            Dgc g2 = dgc[grp][layer][2], c2 = dgc[grp][layer][3];

            // gates
            k_concat<<<(nElemCD + 255) / 256, 256, 0, stream>>>(h_in, h, XH);
            run_dgc(XH, P1, g1, T, G1, 2 * UU, /*sigmoid*/1, stream);
            run_dgc(XH, P2, g2, T, G2, 2 * UU, /*sigmoid*/1, stream);
            // candidates
            k_concat_rh<<<(nElemCD + 255) / 256, 256, 0, stream>>>(h_in, G1, h, XR);
            run_dgc(XR, P1, c1, T, C1, UU, /*tanh*/2, stream);
            k_concat_rh<<<(nElemCD + 255) / 256, 256, 0, stream>>>(h_in, G2, h, XR);
            run_dgc(XR, P2, c2, T, C2, UU, /*tanh*/2, stream);
            // GRU update
            k_update<<<(nElemU + 255) / 256, 256, 0, stream>>>(G1, G2, C1, C2, h);
        }
        if (!is_enc) {
            k_out<<<(BB * NN + 255) / 256, 256, 0, stream>>>(H + (size_t)nElemU,
                Wout, bout, out, GO, step - 12);
        }
    }
}